// VecLoss_12506944766655
// MI455X (gfx1250) — compile-verified
//
#include <hip/hip_runtime.h>
#include <hip/hip_bf16.h>

typedef __attribute__((ext_vector_type(2))) float v2f;
typedef __attribute__((ext_vector_type(8))) float v8f;

#define NPTS 1024          // points per batch
#define NTILE 64           // NPTS / 16
#define THRESH 0.95f

// ws[0] = sum(d2*mask), ws[1] = sum(mask)
__global__ void vl_zero(float* ws) {
    ws[0] = 0.0f;
    ws[1] = 0.0f;
}

// One workgroup = (batch, group of 4 row-tiles). 8 waves: wave w handles
// rowTile = wgRow*4 + w/2 and column tiles [ (w&1)*32, (w&1)*32+32 ).
__global__ __launch_bounds__(256) void vl_main(
    const float* __restrict__ pts,   // (32, 1024, 3)
    const float* __restrict__ sim,   // (32, 1024, 1024)
    float* __restrict__ ws)
{
    __shared__ float spts[NPTS * 3];   // 12 KB
    __shared__ float sx2[NPTS];        //  4 KB

    const int b     = blockIdx.x >> 4;    // batch
    const int wgRow = blockIdx.x & 15;    // row-tile group within batch

    // Stage this batch's points + squared norms into LDS.
    const float* bp = pts + (size_t)b * NPTS * 3;
    for (int p = threadIdx.x; p < NPTS; p += 256) {
        float x = bp[p * 3 + 0];
        float y = bp[p * 3 + 1];
        float z = bp[p * 3 + 2];
        spts[p * 3 + 0] = x;
        spts[p * 3 + 1] = y;
        spts[p * 3 + 2] = z;
        sx2[p] = x * x + y * y + z * z;
    }
    __syncthreads();

    const int wave = threadIdx.x >> 5;       // 0..7
    const int lane = threadIdx.x & 31;
    const int rowTile  = wgRow * 4 + (wave >> 1);
    const int colStart = (wave & 1) * 32;

    // WMMA lane decomposition (wave32, 16x16x4 f32):
    //   A (16x4): lane m = lane%16, K half selected by lane>=16 (K0/K1 vs K2/K3)
    //   B (4x16): lane n = lane%16, same K-half split
    //   C/D (16x16): VGPR v -> row m = v + 8*(lane>=16), col n = lane%16
    const int n    = lane & 15;
    const int half = lane >> 4;       // 0 or 1
    const int mOff = half * 8;
    const int k0   = half * 2;        // K index pair base: 0 or 2

    // A fragment, constant for this wave's row tile; fold the -2 scale in.
    const int arow = rowTile * 16 + n;
    v2f A;
    A.x = -2.0f * spts[arow * 3 + k0];
    A.y = (k0 + 1 < 3) ? (-2.0f * spts[arow * 3 + k0 + 1]) : 0.0f;  // K=3 pad -> 0

    // Row |x|^2 terms for the C preload (rows m = rowTile*16 + mOff + v).
    float cx2[8];
#pragma unroll
    for (int v = 0; v < 8; ++v) cx2[v] = sx2[rowTile * 16 + mOff + v];

    const float* simb = sim + (size_t)b * NPTS * NPTS;
    float accD = 0.0f;   // sum of masked d2
    float accC = 0.0f;   // mask count

    for (int tj = colStart; tj < colStart + 32; ++tj) {
        const int col = tj * 16 + n;
        v2f B;
        B.x = spts[col * 3 + k0];
        B.y = (k0 + 1 < 3) ? spts[col * 3 + k0 + 1] : 0.0f;

        const float colx2 = sx2[col];
        v8f C;
#pragma unroll
        for (int v = 0; v < 8; ++v) C[v] = cx2[v] + colx2;

        // D = (-2*X_rows) * X_cols^T + (x2_i + x2_j) = d2 tile, one instruction.
        v8f D = __builtin_amdgcn_wmma_f32_16x16x4_f32(
            /*neg_a=*/false, A, /*neg_b=*/false, B,
            /*c_mod=*/(short)0, C, /*reuse_a=*/false, /*reuse_b=*/false);

        // Stream the matching 16x16 sim tile (the bandwidth-bound part).
        const float* sp = simb + (size_t)(rowTile * 16 + mOff) * NPTS + tj * 16 + n;
#pragma unroll
        for (int v = 0; v < 8; ++v) {
            float s = sp[(size_t)v * NPTS];
            float d = fmaxf(D[v], 0.0f);
            if (s > THRESH) {
                accD += d;
                accC += 1.0f;
            }
        }
    }

    // Wave32 reduction, then one atomic per wave.
#pragma unroll
    for (int off = 16; off > 0; off >>= 1) {
        accD += __shfl_down(accD, off, 32);
        accC += __shfl_down(accC, off, 32);
    }
    if (lane == 0) {
        atomicAdd(&ws[0], accD);
        atomicAdd(&ws[1], accC);
    }
}

__global__ void vl_finalize(const float* __restrict__ ws, float* __restrict__ out) {
    out[0] = ws[0] / (ws[1] + 1e-8f) * 10.0f;
}

extern "C" void kernel_launch(void* const* d_in, const int* in_sizes, int n_in,
                              void* d_out, int out_size, void* d_ws, size_t ws_size,
                              hipStream_t stream) {
    const float* pts = (const float*)d_in[0];   // (32,1024,3) f32
    const float* sim = (const float*)d_in[1];   // (32,1024,1024) f32
    float* out = (float*)d_out;                 // scalar f32
    float* ws  = (float*)d_ws;                  // 2 floats used

    vl_zero<<<1, 1, 0, stream>>>(ws);
    vl_main<<<32 * 16, 256, 0, stream>>>(pts, sim, ws);
    vl_finalize<<<1, 1, 0, stream>>>(ws, out);
}